// QNNEmbedding_18683107737988
// MI455X (gfx1250) — compile-verified
//
#include <hip/hip_runtime.h>
#include <math.h>

// MI455X / gfx1250, wave32. One wave simulates one batch element's 8-qubit
// state as a 16x16 complex f32 tile; all gate applications are 16x16x16
// complex matmuls built from chained V_WMMA_F32_16X16X4_F32.
//
// Workspace layout (d_ws): float V[6][2][2][256]
//   [layer][mat: 0 = V0 = P_A*R0 (CNOT-perm baked), 1 = R1][plane re/im][r*16+c]
//   = 24576 bytes.

typedef __attribute__((ext_vector_type(2))) float v2f;
typedef __attribute__((ext_vector_type(8))) float v8f;

#define N_LAYERS 6
#define NWAVES   4
#define NTHREADS (NWAVES * 32)

struct c2 { float re, im; };

__device__ __forceinline__ c2 cmul(c2 a, c2 b) {
  return c2{a.re * b.re - a.im * b.im, a.re * b.im + a.im * b.re};
}
// RX(t): [[c, -i s], [-i s, c]]
__device__ __forceinline__ c2 rx_el(int a, int b, float c, float s) {
  return (a == b) ? c2{c, 0.f} : c2{0.f, -s};
}
// RY(t): [[c, -s], [s, c]]
__device__ __forceinline__ c2 ry_el(int a, int b, float c, float s) {
  if (a == b) return c2{c, 0.f};
  return (a == 1) ? c2{s, 0.f} : c2{-s, 0.f};
}

// ---------------------------------------------------------------------------
// Setup: build per-layer shared 16x16 complex matrices from weights.
//   R0 = W0 (x) W1 (x) W2 (x) W3 on wires 0-3 (wire0 = bit3 of index)
//   V0[r,c] = R0[r ^ (r>>1), c]   (Gray perm = CNOT(0,1)(1,2)(2,3) baked in)
//   R1 = same for wires 4-7 (no perm baked; applied as lane permute later)
// where W_i = RZ(w2) RY(w1) RX(w0).
// grid: <<<6, 256>>>
// ---------------------------------------------------------------------------
__global__ void qnn_build_v(const float* __restrict__ wts, float* __restrict__ V) {
  int l = blockIdx.x;          // layer
  int t = threadIdx.x;         // entry 0..255
  int r = t >> 4, c = t & 15;
  for (int m = 0; m < 2; ++m) {
    int rr = (m == 0) ? (r ^ (r >> 1)) : r;   // bake P_A into V0 rows
    c2 prod = c2{1.f, 0.f};
    for (int i = 0; i < 4; ++i) {
      int wire = m * 4 + i;
      float a0 = wts[(l * 8 + wire) * 3 + 0];
      float a1 = wts[(l * 8 + wire) * 3 + 1];
      float a2 = wts[(l * 8 + wire) * 3 + 2];
      float cx = cosf(0.5f * a0), sx = sinf(0.5f * a0);
      float cy = cosf(0.5f * a1), sy = sinf(0.5f * a1);
      float cz = cosf(0.5f * a2), sz = sinf(0.5f * a2);
      int ab = (rr >> (3 - i)) & 1;            // wire i <-> bit (3-i)
      int bb = (c  >> (3 - i)) & 1;
      // (RY*RX)[ab][bb]
      c2 yx = c2{0.f, 0.f};
      for (int k = 0; k < 2; ++k) {
        c2 p = cmul(ry_el(ab, k, cy, sy), rx_el(k, bb, cx, sx));
        yx.re += p.re; yx.im += p.im;
      }
      // RZ = diag(e^{-i a2/2}, e^{+i a2/2})
      c2 z = ab ? c2{cz, sz} : c2{cz, -sz};
      prod = cmul(prod, cmul(z, yx));
    }
    V[(((l * 2 + m) * 2 + 0) * 256) + t] = prod.re;
    V[(((l * 2 + m) * 2 + 1) * 256) + t] = prod.im;
  }
}

// ---------------------------------------------------------------------------
// Complex 16x16 x 16x16 matmul: C = A * B (or A * B^T) with A,B row-major in
// LDS, C in WMMA C-layout registers. 16 v_wmma_f32_16x16x4_f32 per call.
// A-layout: (m,k) -> lane = (m&15) + 16*(k>=2 per 4-block), vgpr = k&1
//   => lane l loads A[l&15][4*kb + 2*(l>>4) + {0,1}]  (contiguous b64)
// B-layout: (k,n) -> lane = n + 16*(k>=2), vgpr = k&1
//   => lane l loads B[4*kb + 2*(l>>4) + {0,1}][l&15]
// B^T from row-major Y: (k,n) = Y[n][k] => same contiguous pattern as A.
// ---------------------------------------------------------------------------
__device__ __forceinline__ void cmm(const float* __restrict__ Ar,
                                    const float* __restrict__ Ai,
                                    const float* __restrict__ Br,
                                    const float* __restrict__ Bi,
                                    bool bT, int l, v8f& Cr, v8f& Ci) {
  int lr = l & 15, lh = l >> 4;
  int aoff = lr * 16 + 2 * lh;
  v8f rr = {}, ii = {}, ri = {};
#pragma unroll
  for (int kb = 0; kb < 4; ++kb) {
    v2f ar = *(const v2f*)(Ar + aoff + 4 * kb);
    v2f ai = *(const v2f*)(Ai + aoff + 4 * kb);
    v2f br, bi;
    if (bT) {
      br = *(const v2f*)(Br + aoff + 4 * kb);
      bi = *(const v2f*)(Bi + aoff + 4 * kb);
    } else {
      int k0 = 4 * kb + 2 * lh;
      br.x = Br[k0 * 16 + lr]; br.y = Br[(k0 + 1) * 16 + lr];
      bi.x = Bi[k0 * 16 + lr]; bi.y = Bi[(k0 + 1) * 16 + lr];
    }
    rr = __builtin_amdgcn_wmma_f32_16x16x4_f32(false, ar, false, br, (short)0, rr, false, false);
    ii = __builtin_amdgcn_wmma_f32_16x16x4_f32(false, ai, false, bi, (short)0, ii, false, false);
    ri = __builtin_amdgcn_wmma_f32_16x16x4_f32(false, ar, false, bi, (short)0, ri, false, false);
    ri = __builtin_amdgcn_wmma_f32_16x16x4_f32(false, ai, false, br, (short)0, ri, false, false);
  }
  Cr = rr - ii;   // f32 WMMA has no A/B negate -> subtract accumulators
  Ci = ri;
}

// C-layout regs -> row-major LDS: vgpr v, lane l holds (m = v + 8*(l>>4), n = l&15)
__device__ __forceinline__ void store_tile(float* __restrict__ dst, int l, v8f c) {
  int base = ((l >> 4) * 8) * 16 + (l & 15);
#pragma unroll
  for (int v = 0; v < 8; ++v) dst[base + v * 16] = c[v];
}

__device__ __forceinline__ float wave_sum(float v, int l) {
#pragma unroll
  for (int m = 16; m; m >>= 1) {
    int src = (l ^ m) << 2;
    v += __int_as_float(__builtin_amdgcn_ds_bpermute(src, __float_as_int(v)));
  }
  return v;
}

// ---------------------------------------------------------------------------
// Main kernel: one wave per batch element.
// ---------------------------------------------------------------------------
__global__ void __launch_bounds__(NTHREADS)
qnn_kernel(const float* __restrict__ x, const float* __restrict__ Vg,
           float* __restrict__ out, int batch) {
  __shared__ float Vl[N_LAYERS * 2 * 2 * 256];     // 24 KB shared matrices
  __shared__ float Ebuf[NWAVES][2][256];           // encoding matrix (per wave)
  __shared__ float Mbuf[NWAVES][2][256];           // M0, reused for T
  __shared__ float Nbuf[NWAVES][2][256];           // N1
  __shared__ float Sbuf[NWAVES][2][256];           // state

  int tid = threadIdx.x;
  int l = tid & 31;        // lane
  int w = tid >> 5;        // wave in block
  int b = blockIdx.x * NWAVES + w;
  int bb = (b < batch) ? b : (batch - 1);

  // cooperative copy of shared V matrices (24KB, L2-resident)
  {
    const float4* src = (const float4*)Vg;
    float4* dst = (float4*)Vl;
    for (int i = tid; i < (N_LAYERS * 2 * 2 * 256) / 4; i += NTHREADS) dst[i] = src[i];
  }
  __syncthreads();

  // Build E = RX(x0) (x) RY(x1) (x) RX(x0) (x) RY(x1)  (same for both groups)
  float x0 = x[2 * bb], x1 = x[2 * bb + 1];
  float c0 = cosf(0.5f * x0), s0 = sinf(0.5f * x0);
  float c1 = cosf(0.5f * x1), s1 = sinf(0.5f * x1);
#pragma unroll
  for (int t = 0; t < 8; ++t) {
    int idx = t * 32 + l;
    int a = idx >> 4, bcol = idx & 15;
    c2 e = rx_el((a >> 3) & 1, (bcol >> 3) & 1, c0, s0);
    e = cmul(e, ry_el((a >> 2) & 1, (bcol >> 2) & 1, c1, s1));
    e = cmul(e, rx_el((a >> 1) & 1, (bcol >> 1) & 1, c0, s0));
    e = cmul(e, ry_el(a & 1, bcol & 1, c1, s1));
    Ebuf[w][0][idx] = e.re;
    Ebuf[w][1][idx] = e.im;
    // init state |0...0>: S[0,0] = 1
    Sbuf[w][0][idx] = (idx == 0) ? 1.f : 0.f;
    Sbuf[w][1][idx] = 0.f;
  }

  // combined CNOT(3,4) + group-1 Gray perm, as lane gather:
  // dst col c' takes src col (c' ^ (c'>>1)) ^ (8 if row odd)
  int csrc = (l & 15); csrc = csrc ^ (csrc >> 1);
  int addrE = (((l & 16) | csrc)) << 2;
  int addrO = addrE ^ (8 << 2);

  for (int layer = 0; layer < N_LAYERS; ++layer) {
    const float* V0r = &Vl[((layer * 2 + 0) * 2 + 0) * 256];
    const float* V0i = &Vl[((layer * 2 + 0) * 2 + 1) * 256];
    const float* R1r = &Vl[((layer * 2 + 1) * 2 + 0) * 256];
    const float* R1i = &Vl[((layer * 2 + 1) * 2 + 1) * 256];
    v8f cr, ci;
    // M0 = V0 * E   (rows: encoding, rotations, then CNOT perm -- baked)
    cmm(V0r, V0i, &Ebuf[w][0][0], &Ebuf[w][1][0], false, l, cr, ci);
    store_tile(&Mbuf[w][0][0], l, cr); store_tile(&Mbuf[w][1][0], l, ci);
    // N1 = R1 * E   (column rotations; perms applied after, in order)
    cmm(R1r, R1i, &Ebuf[w][0][0], &Ebuf[w][1][0], false, l, cr, ci);
    store_tile(&Nbuf[w][0][0], l, cr); store_tile(&Nbuf[w][1][0], l, ci);
    // T = M0 * S
    cmm(&Mbuf[w][0][0], &Mbuf[w][1][0], &Sbuf[w][0][0], &Sbuf[w][1][0], false, l, cr, ci);
    store_tile(&Mbuf[w][0][0], l, cr); store_tile(&Mbuf[w][1][0], l, ci);  // T overwrites M0
    // S2 = T * N1^T
    cmm(&Mbuf[w][0][0], &Mbuf[w][1][0], &Nbuf[w][0][0], &Nbuf[w][1][0], true, l, cr, ci);
    // CNOT(3,4) (row parity == vgpr parity) fused with group-1 CNOT-chain perm
    v8f nr, ni;
#pragma unroll
    for (int v = 0; v < 8; ++v) {
      int ad = (v & 1) ? addrO : addrE;
      nr[v] = __int_as_float(__builtin_amdgcn_ds_bpermute(ad, __float_as_int(cr[v])));
      ni[v] = __int_as_float(__builtin_amdgcn_ds_bpermute(ad, __float_as_int(ci[v])));
    }
    store_tile(&Sbuf[w][0][0], l, nr);
    store_tile(&Sbuf[w][1][0], l, ni);
  }

  // Epilogue: probs and <Z_i>. Row bits: b3=lanehalf(w0), b2..b0 = vgpr(w1..w3).
  int lr = l & 15, lh = l >> 4;
  float p[8];
#pragma unroll
  for (int v = 0; v < 8; ++v) {
    float a = Sbuf[w][0][(v + 8 * lh) * 16 + lr];
    float bi = Sbuf[w][1][(v + 8 * lh) * 16 + lr];
    p[v] = a * a + bi * bi;
  }
  float t  = p[0] + p[1] + p[2] + p[3] + p[4] + p[5] + p[6] + p[7];
  float u1 = p[0] + p[1] + p[2] + p[3] - p[4] - p[5] - p[6] - p[7];  // w1: v bit2
  float u2 = p[0] + p[1] - p[2] - p[3] + p[4] + p[5] - p[6] - p[7];  // w2: v bit1
  float u3 = p[0] - p[1] + p[2] - p[3] + p[4] - p[5] + p[6] - p[7];  // w3: v bit0
  float ev[8];
  ev[0] = lh ? -t : t;                    // w0: lane half
  ev[1] = u1; ev[2] = u2; ev[3] = u3;
  ev[4] = (lr & 8) ? -t : t;              // w4..w7: column bits
  ev[5] = (lr & 4) ? -t : t;
  ev[6] = (lr & 2) ? -t : t;
  ev[7] = (lr & 1) ? -t : t;
#pragma unroll
  for (int i = 0; i < 8; ++i) ev[i] = wave_sum(ev[i], l);

  if (b < batch && l == 0) {
    const float kPi = 3.14159265358979323846f;
#pragma unroll
    for (int i = 0; i < 8; ++i) out[b * 8 + i] = kPi * ev[i];
  }
}

extern "C" void kernel_launch(void* const* d_in, const int* in_sizes, int n_in,
                              void* d_out, int out_size, void* d_ws, size_t ws_size,
                              hipStream_t stream) {
  const float* x   = (const float*)d_in[0];   // [B,2] f32
  const float* wts = (const float*)d_in[1];   // [6,8,3] f32
  float* out = (float*)d_out;                 // [B,8] f32
  float* V   = (float*)d_ws;                  // 24576 bytes used

  int batch = in_sizes[0] / 2;
  qnn_build_v<<<N_LAYERS, 256, 0, stream>>>(wts, V);
  int nblocks = (batch + NWAVES - 1) / NWAVES;
  qnn_kernel<<<nblocks, NTHREADS, 0, stream>>>(x, V, out, batch);
}